// FusedMoEWithLoRA_87582973100132
// MI455X (gfx1250) — compile-verified
//
#include <hip/hip_runtime.h>
#include <hip/hip_bf16.h>

// ---- problem constants (match reference) ----
constexpr int T_  = 1024;
constexpr int H_  = 1024;
constexpr int I_  = 1024;
constexpr int E_  = 8;
constexpr int Kk  = 2;
constexpr int L_  = 4;
constexpr int R_  = 16;
constexpr int N_  = 2 * I_;          // 2048
constexpr int TK  = T_ * Kk;         // 2048 dispatched pairs
constexpr int TKPAD = TK + E_ * 16;  // each expert segment padded to 16 rows
constexpr int ROWTILES = TKPAD / 16; // 136
constexpr int ZPAD = 1024;           // zero-page floats (>= max K-dim row span)

typedef __attribute__((ext_vector_type(2))) float v2f;
typedef __attribute__((ext_vector_type(8))) float v8f;

// V_WMMA_F32_16X16X4_F32 : A 16x4 f32 (2 VGPR), B 4x16 f32 (2 VGPR), C/D 16x16 f32 (8 VGPR)
__device__ __forceinline__ v8f wmma4(v2f a, v2f b, v8f c) {
    return __builtin_amdgcn_wmma_f32_16x16x4_f32(
        /*neg_a=*/false, a, /*neg_b=*/false, b,
        /*c_mod=*/(short)0, c, /*reuse_a=*/false, /*reuse_b=*/false);
}

// ============ routing bookkeeping + zero page ============
__global__ void init_ws_kernel(int* __restrict__ wsi, float* __restrict__ zpad) {
    int idx = blockIdx.x * blockDim.x + threadIdx.x;
    if (idx < 32) wsi[idx] = 0;                // counts[8], cursors[8], offsets[9]
    if (idx < TKPAD) wsi[32 + idx] = -1;       // perm
    if (idx < ZPAD) zpad[idx] = 0.f;           // zero page for dummy A rows
}

__global__ void count_kernel(const int* __restrict__ ids, int* __restrict__ counts) {
    int p = blockIdx.x * blockDim.x + threadIdx.x;
    if (p < TK) atomicAdd(&counts[ids[p]], 1);
}

__global__ void scan_kernel(const int* __restrict__ counts, int* __restrict__ offsets) {
    if (threadIdx.x == 0) {
        int o = 0;
        for (int e = 0; e < E_; e++) {
            offsets[e] = o;
            o += ((counts[e] + 15) >> 4) << 4;   // pad to 16-row tiles
        }
        offsets[E_] = o;
    }
}

__global__ void scatter_kernel(const int* __restrict__ ids, int* __restrict__ cursors,
                               const int* __restrict__ offsets, int* __restrict__ perm) {
    int p = blockIdx.x * blockDim.x + threadIdx.x;
    if (p < TK) {
        int e = ids[p];
        int pos = atomicAdd(&cursors[e], 1);
        perm[offsets[e] + pos] = p;
    }
}

// ============ LoRA-A reduction: tmp[p,r] = sum_d A[l,e,r,d] * src[row,d] ============
__global__ void lora_a_kernel(const float* __restrict__ A,     // [L,E,R,dim]
                              const float* __restrict__ src,   // [*, dim]
                              const int* __restrict__ ids,     // [TK]
                              const int* __restrict__ lor,     // [T]
                              float* __restrict__ tmp,         // [TK, R]
                              int dim, int perTokenSrc) {
    int p = blockIdx.x;
    int t = p / Kk;
    int e = ids[p];
    int l = lor[t];
    const float* Ar = A + (((size_t)l * E_ + e) * R_) * (size_t)dim;
    const float* sr = src + (size_t)(perTokenSrc ? t : p) * dim;
    int tid = threadIdx.x;           // blockDim = 128
    int r = tid & 15, ch = tid >> 4; // 8 chunks
    int chunk = dim >> 3;
    float s = 0.f;
    const float* arow = Ar + (size_t)r * dim;
    for (int d = ch * chunk; d < (ch + 1) * chunk; ++d) s += arow[d] * sr[d];
    __shared__ float red[128];
    red[tid] = s;
    __syncthreads();
    if (tid < 16) {
        float acc = 0.f;
        #pragma unroll
        for (int c = 0; c < 8; c++) acc += red[c * 16 + tid];
        tmp[(size_t)p * R_ + tid] = acc;
    }
}

// ============ gate_up grouped WMMA GEMM + LoRA-B + silu*mul -> act ============
// grid (ROWTILES, I_/512), block 256 (8 waves); each wave: 16 rows x 64 inter-cols
__global__ __launch_bounds__(256) void gateup_wmma_kernel(
    const float* __restrict__ x,        // [T,H]
    const float* __restrict__ w13,      // [E,N,H]
    const float* __restrict__ gB,       // [L,E,N,R]
    const float* __restrict__ scalings, // [L]
    const int* __restrict__ lor,        // [T]
    const int* __restrict__ perm,       // [TKPAD]
    const int* __restrict__ offsets,    // [E+1]
    const float* __restrict__ tmpg,     // [TK,R]
    const float* __restrict__ zpad,     // [ZPAD] zeros
    float* __restrict__ act)            // [TK,I]
{
    __shared__ int   sp[16], slor[16], stok[16];
    __shared__ float sscal[16], stmp[16][16];

    int tileBase = blockIdx.x * 16;
    if (tileBase >= offsets[E_]) return;
    int e = 0;
    #pragma unroll
    for (int i = 1; i < E_; i++) if (tileBase >= offsets[i]) e = i;

    int tid = threadIdx.x;
    if (tid < 16) {
        int p = perm[tileBase + tid];
        sp[tid] = p;
        int tok = p >= 0 ? p / Kk : 0;
        stok[tid] = tok;
        int l = p >= 0 ? lor[tok] : 0;
        slor[tid] = l;
        sscal[tid] = p >= 0 ? scalings[l] : 0.f;
    }
    __syncthreads();
    {
        int row = tid >> 4, r = tid & 15;
        int p = sp[row];
        stmp[row][r] = p >= 0 ? tmpg[(size_t)p * R_ + r] : 0.f;
    }
    __syncthreads();

    int wave = tid >> 5, lane = tid & 31;
    int colbase = (blockIdx.y * 8 + wave) * 64;   // intermediate-dim base
    int rowq = lane & 15;
    int kb = (lane >> 4) * 2;
    // dummy rows read from the zero page: no per-iteration mask VALU needed
    const float* xrow = (sp[rowq] >= 0) ? (x + (size_t)stok[rowq] * H_) : zpad;
    const float* wE = w13 + (size_t)e * N_ * H_;

    v8f accg[4] = {};
    v8f accu[4] = {};
    #pragma unroll 2
    for (int h0 = 0; h0 < H_; h0 += 4) {
        v2f a = *(const v2f*)(xrow + h0 + kb);
        #pragma unroll
        for (int c = 0; c < 4; c++) {
            int ng = colbase + c * 16 + rowq;
            v2f bg = *(const v2f*)(wE + (size_t)ng * H_ + h0 + kb);
            accg[c] = wmma4(a, bg, accg[c]);
            v2f bu = *(const v2f*)(wE + (size_t)(I_ + ng) * H_ + h0 + kb);
            accu[c] = wmma4(a, bu, accu[c]);
        }
    }

    int halfsel = lane >> 4, coll = lane & 15;
    #pragma unroll
    for (int c = 0; c < 4; c++) {
        #pragma unroll
        for (int v = 0; v < 8; v++) {
            int row = v + 8 * halfsel;
            int p = sp[row];
            if (p < 0) continue;
            int i = colbase + c * 16 + coll;
            int l = slor[row];
            const float* bgrow = gB + (((size_t)l * E_ + e) * N_ + i) * R_;
            const float* burow = gB + (((size_t)l * E_ + e) * N_ + (I_ + i)) * R_;
            float dg = 0.f, du = 0.f;
            #pragma unroll
            for (int r = 0; r < R_; r++) {
                float tr = stmp[row][r];
                dg += bgrow[r] * tr;
                du += burow[r] * tr;
            }
            float g = accg[c][v] + sscal[row] * dg;
            float u = accu[c][v] + sscal[row] * du;
            float s = g / (1.f + __expf(-g));   // silu
            act[(size_t)p * I_ + i] = s * u;
        }
    }
}

// ============ down grouped WMMA GEMM -> cache3 (unweighted) ============
// grid (ROWTILES, H_/512), block 256; each wave: 16 rows x 64 hidden-cols
__global__ __launch_bounds__(256) void down_wmma_kernel(
    const float* __restrict__ act,      // [TK,I]
    const float* __restrict__ w2,       // [E,H,I]
    const int* __restrict__ perm,
    const int* __restrict__ offsets,
    const float* __restrict__ zpad,
    float* __restrict__ cache3)         // [TK,H]
{
    __shared__ int sp[16];
    int tileBase = blockIdx.x * 16;
    if (tileBase >= offsets[E_]) return;
    int e = 0;
    #pragma unroll
    for (int i = 1; i < E_; i++) if (tileBase >= offsets[i]) e = i;
    int tid = threadIdx.x;
    if (tid < 16) sp[tid] = perm[tileBase + tid];
    __syncthreads();

    int wave = tid >> 5, lane = tid & 31;
    int colbase = (blockIdx.y * 8 + wave) * 64;
    int rowq = lane & 15;
    int kb = (lane >> 4) * 2;
    int myp = sp[rowq];
    const float* arow = (myp >= 0) ? (act + (size_t)myp * I_) : zpad;
    const float* wE = w2 + (size_t)e * H_ * I_;

    v8f acc[4] = {};
    #pragma unroll 2
    for (int i0 = 0; i0 < I_; i0 += 4) {
        v2f a = *(const v2f*)(arow + i0 + kb);
        #pragma unroll
        for (int c = 0; c < 4; c++) {
            int hc = colbase + c * 16 + rowq;
            v2f b = *(const v2f*)(wE + (size_t)hc * I_ + i0 + kb);
            acc[c] = wmma4(a, b, acc[c]);
        }
    }

    int halfsel = lane >> 4, coll = lane & 15;
    #pragma unroll
    for (int c = 0; c < 4; c++) {
        #pragma unroll
        for (int v = 0; v < 8; v++) {
            int row = v + 8 * halfsel;
            int p = sp[row];
            if (p < 0) continue;
            int h = colbase + c * 16 + coll;
            cache3[(size_t)p * H_ + h] = acc[c][v];
        }
    }
}

// ============ combine: routing weight * (cache3 + down-LoRA-B delta), moe_sum ============
__global__ void combine_kernel(const float* __restrict__ cache3,
                               const float* __restrict__ tmpd,   // [TK,R]
                               const float* __restrict__ dB,     // [L,E,H,R]
                               const float* __restrict__ tw,     // [T,K] flat
                               const float* __restrict__ scalings,
                               const int* __restrict__ ids,      // [TK]
                               const int* __restrict__ lor,      // [T]
                               float* __restrict__ out)          // [T,H]
{
    int idx = blockIdx.x * blockDim.x + threadIdx.x;
    int t = idx / H_;
    int h = idx - t * H_;
    int l = lor[t];
    float sc = scalings[l];
    float acc = 0.f;
    #pragma unroll
    for (int k = 0; k < Kk; k++) {
        int p = t * Kk + k;
        int e = ids[p];
        float wf = tw[p];
        const float* bd = dB + (((size_t)l * E_ + e) * H_ + h) * R_;
        const float* td = tmpd + (size_t)p * R_;
        float d = 0.f;
        #pragma unroll
        for (int r = 0; r < R_; r++) d += bd[r] * td[r];
        acc += wf * (cache3[(size_t)p * H_ + h] + sc * d);
    }
    out[idx] = acc;
}

extern "C" void kernel_launch(void* const* d_in, const int* in_sizes, int n_in,
                              void* d_out, int out_size, void* d_ws, size_t ws_size,
                              hipStream_t stream) {
    const float* x    = (const float*)d_in[0];   // hidden_states [T,H]
    const float* tw   = (const float*)d_in[1];   // topk_weights  [T,K]
    const float* w13  = (const float*)d_in[2];   // [E,N,H]
    const float* w2   = (const float*)d_in[3];   // [E,H,I]
    const float* gA   = (const float*)d_in[4];   // [L,E,R,H]
    const float* gB   = (const float*)d_in[5];   // [L,E,N,R]
    const float* dA   = (const float*)d_in[6];   // [L,E,R,I]
    const float* dB   = (const float*)d_in[7];   // [L,E,H,R]
    const float* scal = (const float*)d_in[8];   // [L]
    const int*   ids  = (const int*)d_in[9];     // [T,K]
    const int*   lor  = (const int*)d_in[10];    // [T]
    float* out = (float*)d_out;

    int* wsi = (int*)d_ws;
    int* counts  = wsi;        // 8
    int* cursors = wsi + 8;    // 8
    int* offsets = wsi + 16;   // 9
    int* perm    = wsi + 32;   // TKPAD

    float* fbase  = (float*)((char*)d_ws + 16384);
    float* zpad   = fbase;                         // ZPAD zeros
    float* tmpg   = zpad + ZPAD;                   // TK*R
    float* tmpd   = tmpg + (size_t)TK * R_;        // TK*R
    float* act    = tmpd + (size_t)TK * R_;        // TK*I
    float* cache3 = act + (size_t)TK * I_;         // TK*H

    init_ws_kernel<<<(TKPAD + 255) / 256, 256, 0, stream>>>(wsi, zpad);
    count_kernel<<<TK / 256, 256, 0, stream>>>(ids, counts);
    scan_kernel<<<1, 32, 0, stream>>>(counts, offsets);
    scatter_kernel<<<TK / 256, 256, 0, stream>>>(ids, cursors, offsets, perm);

    lora_a_kernel<<<TK, 128, 0, stream>>>(gA, x, ids, lor, tmpg, H_, 1);

    gateup_wmma_kernel<<<dim3(ROWTILES, I_ / 512), 256, 0, stream>>>(
        x, w13, gB, scal, lor, perm, offsets, tmpg, zpad, act);

    lora_a_kernel<<<TK, 128, 0, stream>>>(dA, act, ids, lor, tmpd, I_, 0);

    down_wmma_kernel<<<dim3(ROWTILES, H_ / 512), 256, 0, stream>>>(
        act, w2, perm, offsets, zpad, cache3);

    combine_kernel<<<(T_ * H_) / 256, 256, 0, stream>>>(
        cache3, tmpd, dB, tw, scal, ids, lor, out);
}